// SatLossEvaluator_75385265979593
// MI455X (gfx1250) — compile-verified
//
#include <hip/hip_runtime.h>
#include <hip/hip_bf16.h>

typedef __attribute__((ext_vector_type(2))) float v2f;
typedef __attribute__((ext_vector_type(8))) float v8f;

// ---------------------------------------------------------------------------
// Kernel 1: zero workspace (3*F accumulators + 1 sum slot)
// ---------------------------------------------------------------------------
__global__ void sat_zero_ws(float* __restrict__ ws, long n) {
    long i = (long)blockIdx.x * blockDim.x + threadIdx.x;
    long stride = (long)gridDim.x * blockDim.x;
    for (; i < n; i += stride) ws[i] = 0.0f;
}

// ---------------------------------------------------------------------------
// Kernel 2: edge scatter (segment sums via native f32 global atomics)
// ---------------------------------------------------------------------------
__device__ __forceinline__ void edge_one(int v, int f, float ef, float coeff,
                                         const float* __restrict__ vp,
                                         const float* __restrict__ dl,
                                         float* __restrict__ nom,
                                         float* __restrict__ den,
                                         float* __restrict__ tls) {
    float p  = vp[v];
    float ev = ef * p + (1.0f - ef) * 0.5f;
    float w  = expf(coeff * ev);
    unsafeAtomicAdd(&nom[f], w * ev);
    unsafeAtomicAdd(&den[f], w);
    unsafeAtomicAdd(&tls[f], ef * dl[v]);
}

__global__ void sat_edge_scatter(const float* __restrict__ vp,
                                 const float* __restrict__ dl,
                                 const int*   __restrict__ var_idx,
                                 const int*   __restrict__ fun_idx,
                                 const float* __restrict__ ef,
                                 const float* __restrict__ gs,
                                 const float* __restrict__ mc,
                                 float* __restrict__ nom,
                                 float* __restrict__ den,
                                 float* __restrict__ tls,
                                 int e4, int E) {
    const float coeff = fminf(sqrtf(gs[0]), mc[0]);   // min(step^0.5, max_coeff)
    int tid    = blockIdx.x * blockDim.x + threadIdx.x;
    int stride = gridDim.x * blockDim.x;
    for (int i = tid; i < e4; i += stride) {
        int4   v = ((const int4*)  var_idx)[i];
        int4   f = ((const int4*)  fun_idx)[i];
        float4 e = ((const float4*)ef)[i];
        edge_one(v.x, f.x, e.x, coeff, vp, dl, nom, den, tls);
        edge_one(v.y, f.y, e.y, coeff, vp, dl, nom, den, tls);
        edge_one(v.z, f.z, e.z, coeff, vp, dl, nom, den, tls);
        edge_one(v.w, f.w, e.w, coeff, vp, dl, nom, den, tls);
    }
    for (int j = e4 * 4 + tid; j < E; j += stride)    // scalar tail
        edge_one(var_idx[j], fun_idx[j], ef[j], coeff, vp, dl, nom, den, tls);
}

// ---------------------------------------------------------------------------
// Kernel 3: per-clause loss + reduction (WMMA wave-reduce + shuffles)
// ---------------------------------------------------------------------------
__device__ __forceinline__ float clause_log(float n, float d, float t,
                                            float eps, int sharp) {
    float cv = d / fmaxf(n, eps);
    // integer_pow (repeated squaring, matches lax.integer_pow)
    float x = cv - 1.0f, p = 1.0f, base = x;
    int e = sharp;
    while (e > 0) { if (e & 1) p *= base; base *= base; e >>= 1; }
    cv = 1.0f + p;
    cv = t * cv;
    cv = fmaxf(fmaxf(cv, 0.0f), eps);                 // clip(.,0) then max(.,eps)
    return logf(cv);
}

__global__ void sat_clause_reduce(const float* __restrict__ nom,
                                  const float* __restrict__ den,
                                  const float* __restrict__ tls,
                                  const float* __restrict__ eps_p,
                                  const int*   __restrict__ sharp_p,
                                  int f4, int F,
                                  float* __restrict__ sum_out) {
    const float eps  = eps_p[0];
    const int   shrp = sharp_p[0];
    int tid    = blockIdx.x * blockDim.x + threadIdx.x;
    int stride = gridDim.x * blockDim.x;

    float acc = 0.0f;
    for (int i = tid; i < f4; i += stride) {
        float4 n = ((const float4*)nom)[i];
        float4 d = ((const float4*)den)[i];
        float4 t = ((const float4*)tls)[i];
        acc += clause_log(n.x, d.x, t.x, eps, shrp);
        acc += clause_log(n.y, d.y, t.y, eps, shrp);
        acc += clause_log(n.z, d.z, t.z, eps, shrp);
        acc += clause_log(n.w, d.w, t.w, eps, shrp);
    }
    for (int j = f4 * 4 + tid; j < F; j += stride)    // scalar tail
        acc += clause_log(nom[j], den[j], tls[j], eps, shrp);

    // Wave32 reduction, step 1 via WMMA: D[m][n] = colsum(B)[n] = s_n + s_{n+16}
    // A = ones(16x4), B rows 0..1 hold the 32 lane partials (rows 2..3 zero).
    // Executed by every wave with full EXEC (no divergent branch around WMMA).
    v2f a; a[0] = 1.0f; a[1] = 1.0f;
    v2f b; b[0] = acc;  b[1] = 0.0f;
    v8f c = {};
    c = __builtin_amdgcn_wmma_f32_16x16x4_f32(false, a, false, b,
                                              (short)0, c, false, false);
    float t = c[0];                                   // s_{L%16} + s_{L%16+16}
    t += __shfl_xor(t, 1, 32);
    t += __shfl_xor(t, 2, 32);
    t += __shfl_xor(t, 4, 32);
    t += __shfl_xor(t, 8, 32);
    if ((threadIdx.x & 31) == 0)
        unsafeAtomicAdd(sum_out, t);
}

// ---------------------------------------------------------------------------
// Kernel 4: finalize mean
// ---------------------------------------------------------------------------
__global__ void sat_finalize(const float* __restrict__ sum,
                             float* __restrict__ out, float inv_f) {
    out[0] = sum[0] * inv_f;
}

// ---------------------------------------------------------------------------
extern "C" void kernel_launch(void* const* d_in, const int* in_sizes, int n_in,
                              void* d_out, int out_size, void* d_ws, size_t ws_size,
                              hipStream_t stream) {
    const float* vp    = (const float*)d_in[0];   // variable_prediction (V,1)
    const float* dl    = (const float*)d_in[1];   // degree_loss (V,)
    //            label = d_in[2]  (unused by reference math)
    const int*   gm    = (const int*)  d_in[3];   // graph_map (2,E)
    //            bvm   = d_in[4]  (unused)
    const int    F     = in_sizes[5];             // batch_function_map length
    const float* ef    = (const float*)d_in[6];   // edge_feature (E,1)
    //            meta  = d_in[7]  (unused)
    const float* gs    = (const float*)d_in[8];   // global_step
    const float* eps   = (const float*)d_in[9];   // eps
    const float* mc    = (const float*)d_in[10];  // max_coeff
    const int*   sharp = (const int*)  d_in[11];  // loss_sharpness
    const int    E     = in_sizes[6];

    const int* var_idx = gm;
    const int* fun_idx = gm + E;

    float* ws   = (float*)d_ws;
    float* nom  = ws;
    float* den  = ws + (size_t)F;
    float* tls  = ws + 2 * (size_t)F;
    float* sum  = ws + 3 * (size_t)F;

    const long nzero = 3L * F + 1;
    int zblocks = (int)((nzero + 255) / 256);
    if (zblocks > 32768) zblocks = 32768;
    sat_zero_ws<<<zblocks, 256, 0, stream>>>(ws, nzero);

    const int e4 = E / 4;
    int eblocks = (e4 + 255) / 256;
    if (eblocks > 32768) eblocks = 32768;
    if (eblocks < 1) eblocks = 1;
    sat_edge_scatter<<<eblocks, 256, 0, stream>>>(vp, dl, var_idx, fun_idx, ef,
                                                  gs, mc, nom, den, tls, e4, E);

    const int f4 = F / 4;
    int rblocks = (f4 + 255) / 256;
    if (rblocks > 8192) rblocks = 8192;
    if (rblocks < 1) rblocks = 1;
    sat_clause_reduce<<<rblocks, 256, 0, stream>>>(nom, den, tls, eps, sharp,
                                                   f4, F, sum);

    sat_finalize<<<1, 1, 0, stream>>>(sum, (float*)d_out, 1.0f / (float)F);
}